// SoMPredictor_10359461118387
// MI455X (gfx1250) — compile-verified
//
#include <hip/hip_runtime.h>
#include <math.h>

#define HDIM 128
#define GNUM 8192

typedef __attribute__((ext_vector_type(2))) float v2f;
typedef __attribute__((ext_vector_type(8))) float v8f;
typedef unsigned int u32;
typedef u32 u32x4 __attribute__((ext_vector_type(4)));
typedef u32 u32x8 __attribute__((ext_vector_type(8)));

__device__ __forceinline__ float silu_f(float x) {
    return x / (1.0f + __expf(-x));
}

// ---------------- utility kernels ----------------

__global__ void k_zero(float* __restrict__ p, long n) {
    long i = (long)blockIdx.x * blockDim.x + threadIdx.x;
    if (i < n) p[i] = 0.0f;
}

__global__ void k_deg(const int* __restrict__ dst, float* __restrict__ deg, int E) {
    int e = blockIdx.x * blockDim.x + threadIdx.x;
    if (e < E) atomicAdd(&deg[dst[e]], 1.0f);
}

__global__ void k_dis(const float* __restrict__ deg, float* __restrict__ dis, int N) {
    int i = blockIdx.x * blockDim.x + threadIdx.x;
    if (i < N) dis[i] = rsqrtf(deg[i] + 1.0f);
}

__global__ void k_add(float* __restrict__ a, const float* __restrict__ b, long n) {
    long i = (long)blockIdx.x * blockDim.x + threadIdx.x;
    if (i < n) a[i] += b[i];
}

// agg = hw * self_norm + conv_bias   (per-element; self_norm = dis[node]^2)
__global__ void k_init_agg(const float* __restrict__ hw, const float* __restrict__ dis,
                           const float* __restrict__ cb, float* __restrict__ agg, long n) {
    long i = (long)blockIdx.x * blockDim.x + threadIdx.x;
    if (i < n) {
        int node = (int)(i >> 7);
        int k    = (int)(i & 127);
        float s  = dis[node];
        agg[i] = hw[i] * (s * s) + cb[k];
    }
}

// ---------------- WMMA GEMM: out[N,128] = A[N,K] @ W[128,K]^T (+bias)(+acc)(silu) ----
// Block = 256 threads = 8 wave32; wave w owns output columns [16w, 16w+16).
// A tile (16 x K) is DMA'd into LDS by the Tensor Data Mover (one descriptor per
// block, issued by wave 0), with hardware padding of 2 DWORDs per K-DWORD row to
// produce the bank-conflict-free K+2 stride.  tensor_dim1 = n_rows gives free OOB
// zero-fill for the tail tile.  The gather variant (rowidx != nullptr) falls back
// to a cooperative VMEM copy.  B fragments are read from the tiny L2-resident
// weight matrix.  Fragment packing per CDNA5 ISA 7.12.2; EXEC is all-ones on every
// WMMA (no divergence before the stores).

__global__ __launch_bounds__(256)
void k_gemm128(const float* __restrict__ A, int lda,
               const int* __restrict__ rowidx,
               const float* __restrict__ W, int ldw,
               const float* __restrict__ bias,
               float* __restrict__ out,
               int K, int n_rows, int accum, int act)
{
    __shared__ __align__(16) float As[16 * 130];   // K <= 128, stride K+2 (pad)

    const int tid  = threadIdx.x;
    const int lane = tid & 31;
    const int wave = tid >> 5;
    const int r0   = blockIdx.x << 4;
    const int ldk  = K + 2;

    if (rowidx == nullptr) {
        // ---- Tensor Data Mover path: DMA 16 x K tile (row stride lda) into LDS ----
        if (wave == 0) {
            unsigned long long ga =
                (unsigned long long)(const void*)(A + (long)r0 * lda);
            u32 lds_base = (u32)(unsigned long long)(const void*)&As[0];
            // pad_interval: DWORDs between pads = 2^(code+1); K=128 -> 6, K=64 -> 5
            u32 pi = (K == 128) ? 6u : 5u;

            u32x4 g0;
            g0.x = 1u;                                   // count=1, user mode, no gather
            g0.y = lds_base;                             // D#.lds_addr
            g0.z = (u32)ga;                              // global_addr[31:0]
            g0.w = (u32)((ga >> 32) & 0x1FFFFFFu)        // global_addr[56:32]
                 | (2u << 30);                           // type = 2 ("image")

            u32x8 g1;
            g1.s0 = (2u << 16)                           // data_size = 4 bytes
                  | (1u << 20)                           // pad_enable
                  | (pi << 22)                           // pad_interval
                  | (1u << 25);                          // pad_amount = 2 DWORDs
            g1.s1 = ((u32)lda & 0xFFFFu) << 16;          // tensor_dim0[15:0]
            g1.s2 = ((u32)lda >> 16)                     // tensor_dim0[31:16]
                  | (((u32)n_rows & 0xFFFFu) << 16);     // tensor_dim1[15:0]
            g1.s3 = ((u32)n_rows >> 16)                  // tensor_dim1[31:16]
                  | ((u32)K << 16);                      // tile_dim0 = K
            g1.s4 = 16u;                                 // tile_dim1 = 16 rows
            g1.s5 = (u32)lda;                            // tensor_dim0_stride[31:0]
            g1.s6 = 0u;                                  // stride hi / dim1_stride lo
            g1.s7 = 0u;

            asm volatile("tensor_load_to_lds %0, %1"
                         :: "s"(g0), "s"(g1)
                         : "memory");
            __builtin_amdgcn_s_wait_tensorcnt(0);
        }
    } else {
        // ---- gather path: cooperative load of 16 x K tile via rowidx ----
        for (int i = tid; i < (K << 4); i += 256) {
            int r = i / K;
            int c = i - r * K;
            int row = r0 + r;
            if (row >= n_rows) row = n_rows - 1;
            int arow = rowidx[row];
            As[r * ldk + c] = A[(long)arow * lda + c];
        }
    }
    __syncthreads();

    const int c0 = wave << 4;                 // column tile base (8 waves x 16 = 128)
    const int m  = lane & 15;                 // A row within tile / B column within tile
    const int h2 = (lane >> 4) << 1;          // K sub-offset: lanes 16-31 hold K+2,K+3

    const float* wrow = W + (long)(c0 + m) * ldw;
    __builtin_prefetch(wrow, 0, 3);           // global_prefetch_b8 (gfx1250)

    v8f c = {};
    for (int kk = 0; kk < K; kk += 4) {
        // A fragment: lane L -> A[m][kk + 2*(L>>4) + {0,1}]
        v2f a = *(const v2f*)(&As[m * ldk + kk + h2]);
        // B fragment: lane L -> B[kk + 2*(L>>4) + {0,1}][n] = W[n][same K idx]
        v2f b = *(const v2f*)(&wrow[kk + h2]);
        c = __builtin_amdgcn_wmma_f32_16x16x4_f32(
                /*neg_a=*/false, a, /*neg_b=*/false, b,
                /*c_mod=*/(short)0, c, /*reuse_a=*/false, /*reuse_b=*/false);
    }

    // C/D layout: VGPR v, lanes 0-15 -> M=v, lanes 16-31 -> M=v+8; N = lane&15
    const int half8 = (lane >> 4) << 3;
    const int col   = c0 + m;
    const float bv  = bias ? bias[col] : 0.0f;
#pragma unroll
    for (int v = 0; v < 8; ++v) {
        int row = r0 + v + half8;
        if (row < n_rows) {
            long idx  = (long)row * HDIM + col;
            float val = c[v] + bv;
            if (accum) val += out[idx];
            if (act)   val = silu_f(val);
            out[idx] = val;
        }
    }
}

// ---------------- edge scatter: agg[dst] += hw[src] * dis[src]*dis[dst] ------------
// one wave32 per edge; float4 per lane = one full 128-float row per wave

__global__ __launch_bounds__(256)
void k_scatter(const float* __restrict__ hw, const int* __restrict__ src,
               const int* __restrict__ dst, const float* __restrict__ dis,
               float* __restrict__ agg, int E)
{
    int wid  = (int)(((long)blockIdx.x * blockDim.x + threadIdx.x) >> 5);
    int lane = threadIdx.x & 31;
    if (wid >= E) return;
    int s = src[wid], d = dst[wid];
    float nrm = dis[s] * dis[d];
    __builtin_prefetch(agg + (long)d * HDIM, 0, 1);   // warm dest row toward L2
    float4 v = ((const float4*)(hw + (long)s * HDIM))[lane];
    float* ar = agg + (long)d * HDIM + lane * 4;
    atomicAdd(ar + 0, v.x * nrm);
    atomicAdd(ar + 1, v.y * nrm);
    atomicAdd(ar + 2, v.z * nrm);
    atomicAdd(ar + 3, v.w * nrm);
}

// ---------------- fused SiLU + LayerNorm + residual (wave per node, wave32) --------

__global__ __launch_bounds__(256)
void k_fuse(const float* __restrict__ agg, const float* __restrict__ g,
            const float* __restrict__ b, float* __restrict__ h, int N)
{
    int wid  = (int)(((long)blockIdx.x * blockDim.x + threadIdx.x) >> 5);
    int lane = threadIdx.x & 31;
    if (wid >= N) return;
    float4 a = ((const float4*)(agg + (long)wid * HDIM))[lane];
    float s0 = silu_f(a.x), s1 = silu_f(a.y), s2 = silu_f(a.z), s3 = silu_f(a.w);
    float sum = s0 + s1 + s2 + s3;
#pragma unroll
    for (int off = 16; off > 0; off >>= 1) sum += __shfl_xor(sum, off, 32);
    float mu = sum * (1.0f / 128.0f);
    float d0 = s0 - mu, d1 = s1 - mu, d2 = s2 - mu, d3 = s3 - mu;
    float vs = d0 * d0 + d1 * d1 + d2 * d2 + d3 * d3;
#pragma unroll
    for (int off = 16; off > 0; off >>= 1) vs += __shfl_xor(vs, off, 32);
    float rs = rsqrtf(vs * (1.0f / 128.0f) + 1e-5f);
    int k = lane * 4;
    float* hr = h + (long)wid * HDIM + k;
    hr[0] += d0 * rs * g[k + 0] + b[k + 0];
    hr[1] += d1 * rs * g[k + 1] + b[k + 1];
    hr[2] += d2 * rs * g[k + 2] + b[k + 2];
    hr[3] += d3 * rs * g[k + 3] + b[k + 3];
}

// ---------------- global mean pool ----------------

__global__ __launch_bounds__(256)
void k_pool(const float* __restrict__ h, const int* __restrict__ batch,
            float* __restrict__ gsum, float* __restrict__ gcnt, int N)
{
    int wid  = (int)(((long)blockIdx.x * blockDim.x + threadIdx.x) >> 5);
    int lane = threadIdx.x & 31;
    if (wid >= N) return;
    int g = batch[wid];
    float4 v = ((const float4*)(h + (long)wid * HDIM))[lane];
    float* gr = gsum + (long)g * HDIM + lane * 4;
    atomicAdd(gr + 0, v.x);
    atomicAdd(gr + 1, v.y);
    atomicAdd(gr + 2, v.z);
    atomicAdd(gr + 3, v.w);
    if (lane == 0) atomicAdd(&gcnt[g], 1.0f);
}

__global__ void k_div(float* __restrict__ gsum, const float* __restrict__ gcnt, long n) {
    long i = (long)blockIdx.x * blockDim.x + threadIdx.x;
    if (i < n) gsum[i] /= fmaxf(gcnt[i >> 7], 1.0f);
}

// ---------------- final head: out[n] = dot(hid[n], w2) + b2 -----------------------

__global__ __launch_bounds__(256)
void k_head2(const float* __restrict__ hid, const float* __restrict__ w2,
             const float* __restrict__ b2, float* __restrict__ out, int N)
{
    int wid  = (int)(((long)blockIdx.x * blockDim.x + threadIdx.x) >> 5);
    int lane = threadIdx.x & 31;
    if (wid >= N) return;
    float4 v = ((const float4*)(hid + (long)wid * HDIM))[lane];
    float4 w = ((const float4*)w2)[lane];
    float p = v.x * w.x + v.y * w.y + v.z * w.z + v.w * w.w;
#pragma unroll
    for (int off = 16; off > 0; off >>= 1) p += __shfl_xor(p, off, 32);
    if (lane == 0) out[wid] = p + b2[0];
}

// ---------------- host orchestration ----------------

extern "C" void kernel_launch(void* const* d_in, const int* in_sizes, int n_in,
                              void* d_out, int out_size, void* d_ws, size_t ws_size,
                              hipStream_t stream)
{
    const float* x      = (const float*)d_in[0];
    const int*   ei     = (const int*)  d_in[1];
    const int*   batch  = (const int*)  d_in[2];
    const float* in_w   = (const float*)d_in[3];
    const float* in_b   = (const float*)d_in[4];
    const float* conv_w = (const float*)d_in[5];
    const float* conv_b = (const float*)d_in[6];
    const float* ln_g   = (const float*)d_in[7];
    const float* ln_b   = (const float*)d_in[8];
    const float* pw1    = (const float*)d_in[9];
    const float* pb1    = (const float*)d_in[10];
    const float* pw2    = (const float*)d_in[11];
    const float* pb2    = (const float*)d_in[12];
    const float* hw1    = (const float*)d_in[13];
    const float* hb1    = (const float*)d_in[14];
    const float* hw2    = (const float*)d_in[15];
    const float* hb2    = (const float*)d_in[16];

    const int N   = in_sizes[2];
    const int E   = in_sizes[1] / 2;
    const int DIN = in_sizes[3] / HDIM;      // 64
    const int L   = in_sizes[6] / HDIM;      // 4
    const int G   = GNUM;

    // workspace layout (floats)
    float* ws   = (float*)d_ws;
    float* bufH = ws;                              // [N,128] node features h
    float* bufT = bufH + (size_t)N * HDIM;         // [N,128] hw / tmp
    float* bufU = bufT + (size_t)N * HDIM;         // [N,128] agg / tmp
    float* deg  = bufU + (size_t)N * HDIM;         // [N]
    float* dis  = deg + N;                         // [N]
    float* gsum = dis + N;                         // [G,128]
    float* gcnt = gsum + (size_t)G * HDIM;         // [G]

    const int* srcI = ei;
    const int* dstI = ei + E;

    dim3 blk(256);
    const long NH = (long)N * HDIM;

    // zero deg..gcnt span (deg, dis, gsum, gcnt are contiguous)
    long zlen = (long)2 * N + (long)G * HDIM + G;
    k_zero<<<dim3((unsigned)((zlen + 255) / 256)), blk, 0, stream>>>(deg, zlen);

    // degrees and deg^{-1/2}
    k_deg<<<dim3((E + 255) / 256), blk, 0, stream>>>(dstI, deg, E);
    k_dis<<<dim3((N + 255) / 256), blk, 0, stream>>>(deg, dis, N);

    dim3 ggrid((N + 15) / 16);

    // h = x @ in_w^T + in_b
    k_gemm128<<<ggrid, blk, 0, stream>>>(x, DIN, nullptr, in_w, DIN, in_b,
                                         bufH, DIN, N, 0, 0);

    for (int l = 0; l < L; ++l) {
        // hw = h @ conv_w[l]^T
        k_gemm128<<<ggrid, blk, 0, stream>>>(bufH, HDIM, nullptr,
                                             conv_w + (size_t)l * HDIM * HDIM, HDIM,
                                             nullptr, bufT, HDIM, N, 0, 0);
        // agg = hw * self_norm + conv_b[l]
        k_init_agg<<<dim3((unsigned)((NH + 255) / 256)), blk, 0, stream>>>(
            bufT, dis, conv_b + (size_t)l * HDIM, bufU, NH);
        // agg[dst] += hw[src] * norm_e
        k_scatter<<<dim3((E + 7) / 8), blk, 0, stream>>>(bufT, srcI, dstI, dis, bufU, E);
        // h += LayerNorm(silu(agg)) * g + b
        k_fuse<<<dim3((N + 7) / 8), blk, 0, stream>>>(
            bufU, ln_g + (size_t)l * HDIM, ln_b + (size_t)l * HDIM, bufH, N);
    }

    // graph mean pool -> gsum becomes graph_emb in place
    k_pool<<<dim3((N + 7) / 8), blk, 0, stream>>>(bufH, batch, gsum, gcnt, N);
    long gh = (long)G * HDIM;
    k_div<<<dim3((unsigned)((gh + 255) / 256)), blk, 0, stream>>>(gsum, gcnt, gh);

    // phys MLP: bufT = silu(h @ pw1^T + pb1); bufU = bufT @ pw2^T + pb2; bufU += h
    k_gemm128<<<ggrid, blk, 0, stream>>>(bufH, HDIM, nullptr, pw1, HDIM, pb1,
                                         bufT, HDIM, N, 0, 1);
    k_gemm128<<<ggrid, blk, 0, stream>>>(bufT, HDIM, nullptr, pw2, HDIM, pb2,
                                         bufU, HDIM, N, 0, 0);
    k_add<<<dim3((unsigned)((NH + 255) / 256)), blk, 0, stream>>>(bufU, bufH, NH);

    // head layer 1, split over the 2H concat:
    //   bufT  = (h+phys) @ W1[:, :H]^T
    //   bufT  = silu(bufT + gemb[batch] @ W1[:, H:]^T + b1)
    k_gemm128<<<ggrid, blk, 0, stream>>>(bufU, HDIM, nullptr, hw1, 2 * HDIM, nullptr,
                                         bufT, HDIM, N, 0, 0);
    k_gemm128<<<ggrid, blk, 0, stream>>>(gsum, HDIM, batch, hw1 + HDIM, 2 * HDIM, hb1,
                                         bufT, HDIM, N, 1, 1);

    // head layer 2: out[n] = dot(bufT[n], hw2) + hb2
    k_head2<<<dim3((N + 7) / 8), blk, 0, stream>>>(bufT, hw2, hb2, (float*)d_out, N);

    (void)n_in; (void)out_size; (void)ws_size;
}